// QuickFarthestPointSampling_38001870635078
// MI455X (gfx1250) — compile-verified
//
#include <hip/hip_runtime.h>

// Farthest Point Sampling, exact, for MI455X (gfx1250).
// points (32, 65536, 3) f32 -> indices (32,1024) i32, sampled (32,1024,3) f32,
// concatenated flat into d_out in return order.
//
// Design (reasoned from MI455X hardware):
//  - Dataset = 24 MB -> fully L2-resident (192 MB); HBM touched once (~1 us @ 23.3 TB/s).
//  - 1023 strictly serial steps; each must touch all P running-min distances.
//  - min_d[65536] (256 KB) lives in LDS for the whole kernel — a CDNA5-only move
//    (320 KB/WGP vs CDNA4's 64 KB/CU). Only the point stream hits L2 (768 KB/step/WG).
//  - One 32-wave workgroup per batch fills one WGP; wave32 shuffle argmax with
//    first-occurrence tie-break (matches jnp.argmax).
//  - No WMMA: rank-1 update, K=3, serial dependence -> zero column reuse for the
//    matrix pipes; kernel is L2-stream + serial-latency bound, not FLOP bound.
//  - No TDM/async-LDS staging: TENSORcnt/ASYNCcnt are per-wave but tiles would be
//    workgroup-shared -> one extra barrier per tile; min_d owns the LDS budget anyway.
//  - global_prefetch_b8 (gfx1250) pulls the point stream ~48 KB ahead into WGP$.

#define FPS_B        32
#define FPS_P        65536
#define FPS_NS       1024
#define FPS_THREADS  1024                    // 32 waves (wave32)
#define FPS_WAVES    (FPS_THREADS / 32)      // 32
#define FPS_PPT      (FPS_P / FPS_THREADS)   // 64 points per thread

// LDS: min_d[P] (256 KB) | q[4] | redV[WAVES] | redI[WAVES]
#define FPS_LDS_BYTES ((size_t)FPS_P * 4 + 16 + FPS_WAVES * 4 + FPS_WAVES * 4)

extern "C" __global__ __launch_bounds__(FPS_THREADS, 1)
void fps_persistent_kernel(const float* __restrict__ points,
                           int*   __restrict__ outIdx,
                           float* __restrict__ outPts)
{
    extern __shared__ unsigned char smem[];
    float* __restrict__ minD = reinterpret_cast<float*>(smem);                     // P floats
    float* __restrict__ qxyz = reinterpret_cast<float*>(smem + (size_t)FPS_P * 4); // 4 floats
    float* __restrict__ redV = reinterpret_cast<float*>(smem + (size_t)FPS_P * 4 + 16);
    int*   __restrict__ redI = reinterpret_cast<int*>  (smem + (size_t)FPS_P * 4 + 16 + FPS_WAVES * 4);

    const int b   = blockIdx.x;
    const int tid = threadIdx.x;

    const float* __restrict__ pts = points + (size_t)b * FPS_P * 3;
    int*   __restrict__ oIdx = outIdx + (size_t)b * FPS_NS;
    float* __restrict__ oPts = outPts + (size_t)b * FPS_NS * 3;

    // ---- init: min_d = BIG; first selected index is 0 (matches reference) ----
    #pragma unroll
    for (int k = 0; k < FPS_PPT; ++k)
        minD[k * FPS_THREADS + tid] = 1e10f;

    if (tid == 0) {
        const float x = pts[0], y = pts[1], z = pts[2];
        qxyz[0] = x; qxyz[1] = y; qxyz[2] = z;
        oIdx[0] = 0;
        oPts[0] = x; oPts[1] = y; oPts[2] = z;
    }
    __syncthreads();

    // ---- 1023 serial selection steps ----
    for (int s = 1; s < FPS_NS; ++s) {
        const float qx = qxyz[0];
        const float qy = qxyz[1];
        const float qz = qxyz[2];

        float bestV = -1.0f;
        int   bestI = 0;

        // Sweep: strided partition -> each wave's load covers 384 contiguous bytes
        // (global_load_b96, coalesced) and LDS min_d access is conflict-free
        // (stride-1 dwords across lanes). Unroll 16 => 16 loads in flight per wave.
        #pragma unroll 16
        for (int k = 0; k < FPS_PPT; ++k) {
            const int i = k * FPS_THREADS + tid;

            // gfx1250 global_prefetch_b8: stay ~4 chunks (~48 KB) ahead — within
            // the WGP cache working set, issued once per 4 unrolled steps.
            if ((k & 3) == 0 && (k + 4) < FPS_PPT)
                __builtin_prefetch(pts + (size_t)(k + 4) * FPS_THREADS * 3 + (size_t)tid * 3, 0, 1);

            const float* pp = pts + (size_t)i * 3;
            const float px = pp[0];
            const float py = pp[1];
            const float pz = pp[2];

            const float dx = px - qx;
            const float dy = py - qy;
            const float dz = pz - qz;
            const float d  = fmaf(dz, dz, fmaf(dy, dy, dx * dx));

            float m = minD[i];
            m = fminf(m, d);
            minD[i] = m;

            // strictly-greater keeps the smallest index within this lane
            if (m > bestV) { bestV = m; bestI = i; }
        }

        // ---- stage 1: wave32 shuffle argmax (first-occurrence tie-break) ----
        #pragma unroll
        for (int off = 16; off > 0; off >>= 1) {
            const float ov = __shfl_down(bestV, off, 32);
            const int   oi = __shfl_down(bestI, off, 32);
            if (ov > bestV || (ov == bestV && oi < bestI)) { bestV = ov; bestI = oi; }
        }

        const int wave = tid >> 5;
        if ((tid & 31) == 0) { redV[wave] = bestV; redI[wave] = bestI; }
        __syncthreads();

        // ---- stage 2: wave 0 reduces the 32 wave-partials in parallel ----
        if (tid < 32) {
            float bv = redV[tid];
            int   bi = redI[tid];
            #pragma unroll
            for (int off = 16; off > 0; off >>= 1) {
                const float ov = __shfl_down(bv, off, 32);
                const int   oi = __shfl_down(bi, off, 32);
                if (ov > bv || (ov == bv && oi < bi)) { bv = ov; bi = oi; }
            }
            if (tid == 0) {
                // Winner's cache line was touched during the sweep -> WGP$ hit.
                const float* sp = pts + (size_t)bi * 3;
                const float x = sp[0], y = sp[1], z = sp[2];
                qxyz[0] = x; qxyz[1] = y; qxyz[2] = z;
                oIdx[s] = bi;
                oPts[(size_t)s * 3 + 0] = x;
                oPts[(size_t)s * 3 + 1] = y;
                oPts[(size_t)s * 3 + 2] = z;
            }
        }
        __syncthreads();
    }
}

extern "C" void kernel_launch(void* const* d_in, const int* in_sizes, int n_in,
                              void* d_out, int out_size, void* d_ws, size_t ws_size,
                              hipStream_t stream)
{
    (void)in_sizes; (void)n_in; (void)d_ws; (void)ws_size; (void)out_size;

    const float* points = reinterpret_cast<const float*>(d_in[0]);

    // d_out layout (flat, return order): [ indices: 32*1024 i32 | sampled: 32*1024*3 f32 ]
    int*   outIdx = reinterpret_cast<int*>(d_out);
    float* outPts = reinterpret_cast<float*>(d_out) + (size_t)FPS_B * FPS_NS;

    // Opt in to >64 KB dynamic LDS (CDNA5 WGP: 320 KB). Idempotent, capture-safe.
    hipFuncSetAttribute(reinterpret_cast<const void*>(fps_persistent_kernel),
                        hipFuncAttributeMaxDynamicSharedMemorySize,
                        (int)FPS_LDS_BYTES);

    fps_persistent_kernel<<<dim3(FPS_B), dim3(FPS_THREADS), FPS_LDS_BYTES, stream>>>(
        points, outIdx, outPts);
}